// UpsampleLoss_9749575762866
// MI455X (gfx1250) — compile-verified
//
#include <hip/hip_runtime.h>
#include <hip/hip_bf16.h>
#include <math.h>

typedef __attribute__((ext_vector_type(2))) float v2f;
typedef __attribute__((ext_vector_type(8))) float v8f;

#define NPTS      1024
#define NB        4
#define NN_SIZE   5
#define RADIUS_C  0.07f
#define H_C       0.03f
#define EPS_C     1e-12f
#define AUC_EPS   0.01f
#define MAX_ITERS 2000

// workspace layout (floats)
//  [0, 4*1024*1024)          : D2T, transposed distance matrix  D2T[b][j][i] = ||pred_i - gt_j||^2
//  [4194304, 4194308)        : per-batch dist2 (emd partial)
//  [4194308, 4194324)        : 16 rep partials
#define WS_D2T    0
#define WS_DIST2  (NB * NPTS * NPTS)
#define WS_REP    (WS_DIST2 + NB)

// ---------------------------------------------------------------------------
// Kernel 1: pairwise squared distances pred vs gt via V_WMMA_F32_16X16X4_F32.
// One wave computes one 16x16 tile of dot products (K=3 padded to 4), then
// d2 = aa + bb - 2*ab in VALU, stored transposed for coalesced auction scans.
// ---------------------------------------------------------------------------
__global__ __launch_bounds__(256)
void d2_wmma_kernel(const float* __restrict__ pred,
                    const float* __restrict__ gt,
                    float* __restrict__ ws) {
    __shared__ float aaS[8][16];

    const int lane = threadIdx.x & 31;
    const int wv   = threadIdx.x >> 5;
    const int w    = blockIdx.x * 8 + wv;          // global wave = tile id
    const int b    = w >> 12;                      // 4096 tiles per batch
    const int t    = w & 4095;
    const int i0   = (t >> 6) << 4;                // pred-row tile origin
    const int j0   = (t & 63) << 4;                // gt-row (column) tile origin

    const float* pb = pred + b * NPTS * 3;
    const float* gb = gt   + b * NPTS * 3;

    // A matrix 16x4 f32: lanes 0-15 hold K=0,1 ; lanes 16-31 hold K=2,3 (K=3 pad 0)
    const int mh = lane & 15;
    v2f a;
    if (lane < 16) {
        a.x = pb[(i0 + mh) * 3 + 0];
        a.y = pb[(i0 + mh) * 3 + 1];
    } else {
        a.x = pb[(i0 + mh) * 3 + 2];
        a.y = 0.0f;
    }
    // B matrix 4x16 (K x N), B[k][n] = gt[j0+n][k]; lanes 0-15: K=0,1 ; 16-31: K=2,3
    v2f bm;
    if (lane < 16) {
        bm.x = gb[(j0 + mh) * 3 + 0];
        bm.y = gb[(j0 + mh) * 3 + 1];
    } else {
        bm.x = gb[(j0 + mh) * 3 + 2];
        bm.y = 0.0f;
    }

    v8f c = {};
    c = __builtin_amdgcn_wmma_f32_16x16x4_f32(
            /*neg_a=*/false, a, /*neg_b=*/false, bm,
            /*c_mod=*/(short)0, c, /*reuse_a=*/false, /*reuse_b=*/false);

    // stage row norms aa[m] for this wave's tile
    if (lane < 16) {
        float x = pb[(i0 + lane) * 3 + 0];
        float y = pb[(i0 + lane) * 3 + 1];
        float z = pb[(i0 + lane) * 3 + 2];
        aaS[wv][lane] = x * x + y * y + z * z;
    }
    __syncthreads();

    // column norm for this lane's N
    const int n = lane & 15;
    {
        float gx = gb[(j0 + n) * 3 + 0];
        float gy = gb[(j0 + n) * 3 + 1];
        float gz = gb[(j0 + n) * 3 + 2];
        float bb = gx * gx + gy * gy + gz * gz;
        const int moff = (lane < 16) ? 0 : 8;

        float o[8];
        #pragma unroll
        for (int r = 0; r < 8; ++r) {
            float d2 = aaS[wv][moff + r] + bb - 2.0f * c[r];
            o[r] = d2 < 0.0f ? 0.0f : d2;
        }
        // transposed store: D2T[b][j0+n][i0+moff .. +7]  (contiguous -> 2x b128)
        float* dst = ws + WS_D2T + ((size_t)b * NPTS + (j0 + n)) * NPTS + i0 + moff;
        *(float4*)(dst)     = make_float4(o[0], o[1], o[2], o[3]);
        *(float4*)(dst + 4) = make_float4(o[4], o[5], o[6], o[7]);
    }
}

// ---------------------------------------------------------------------------
// Kernel 2: Jacobi auction, one workgroup (1024 threads = 32 waves) per batch.
// State in LDS; scatter-max via ds_max_u64 with (bid_bits<<32)|person key,
// which reproduces the reference's max-bid / max-person tie-break exactly.
// Ends with a deterministic LDS tree reduction for the EMD term.
// ---------------------------------------------------------------------------
__global__ __launch_bounds__(1024)
void auction_kernel(const float* __restrict__ pred,
                    const float* __restrict__ gt,
                    float* __restrict__ ws) {
    __shared__ float              price[NPTS];
    __shared__ int                p2o[NPTS];
    __shared__ int                o2p[NPTS];
    __shared__ unsigned long long bid64[NPTS];
    __shared__ float              red[NPTS];
    __shared__ unsigned           cnt;

    const int tid = threadIdx.x;
    const int b   = blockIdx.x;
    const float* d2row = ws + WS_D2T + (size_t)b * NPTS * NPTS;  // D2T[b][j][i]

    price[tid] = 0.0f;
    p2o[tid]   = -1;
    o2p[tid]   = -1;
    __syncthreads();

    for (int it = 0; it < MAX_ITERS; ++it) {
        if (tid == 0) cnt = 0u;
        bid64[tid] = 0ull;
        __syncthreads();
        if (p2o[tid] < 0) atomicAdd(&cnt, 1u);
        __syncthreads();
        if (cnt == 0u) break;

        if (p2o[tid] < 0) {
            // top-2 of val[j] = -d2(tid,j) - price[j]; strict '>' keeps the
            // smallest index on ties (matches lax.top_k stability).
            float v1 = -INFINITY, v2 = -INFINITY;
            int   i1 = 0;
            for (int j = 0; j < NPTS; ++j) {
                float v = -d2row[(size_t)j * NPTS + tid] - price[j];
                if (v > v1)      { v2 = v1; v1 = v; i1 = j; }
                else if (v > v2) { v2 = v; }
            }
            float bid = price[i1] + (v1 - v2) + AUC_EPS;   // > 0 always
            unsigned long long key =
                ((unsigned long long)__float_as_uint(bid) << 32) |
                (unsigned long long)(unsigned)tid;
            atomicMax(&bid64[i1], key);                    // ds_max_u64
        }
        __syncthreads();

        // object tid resolves its winner
        unsigned long long k = bid64[tid];
        if (k != 0ull) {
            int   winner = (int)(unsigned)(k & 0xffffffffull);
            float bp     = __uint_as_float((unsigned)(k >> 32));
            int   old    = o2p[tid];
            if (old >= 0) p2o[old] = -1;   // old was assigned, winner was not -> no conflict
            p2o[winner] = tid;
            o2p[tid]    = winner;
            price[tid]  = bp;
        }
        __syncthreads();
    }
    __syncthreads();

    // EMD term: sum over points of ||pred - gt[idx]||^2, deterministic tree sum
    const float* pb = pred + b * NPTS * 3;
    const float* gb = gt   + b * NPTS * 3;
    int idx = p2o[tid] < 0 ? 0 : p2o[tid];
    float dx = pb[tid * 3 + 0] - gb[idx * 3 + 0];
    float dy = pb[tid * 3 + 1] - gb[idx * 3 + 1];
    float dz = pb[tid * 3 + 2] - gb[idx * 3 + 2];
    red[tid] = dx * dx + dy * dy + dz * dz;
    __syncthreads();
    for (int s = NPTS / 2; s > 0; s >>= 1) {
        if (tid < s) red[tid] += red[tid + s];
        __syncthreads();
    }
    if (tid == 0) ws[WS_DIST2 + b] = red[0] / (float)(NPTS * 3);
}

// ---------------------------------------------------------------------------
// Kernel 3: KNN repulsion. 16 blocks x 256 threads; pred batch staged in LDS.
// Per-point stable top-5 smallest d2; neighbors 1..4 contribute
// (RADIUS - sqrt(g2)) * exp(-g2/H^2). Deterministic block reduction.
// ---------------------------------------------------------------------------
__global__ __launch_bounds__(256)
void rep_kernel(const float* __restrict__ pred, float* __restrict__ ws) {
    __shared__ float P[NPTS * 3];
    __shared__ float red[256];

    const int tid = threadIdx.x;
    const int b   = blockIdx.x >> 2;
    const int p0  = (blockIdx.x & 3) * 256;
    const float* pb = pred + b * NPTS * 3;

    for (int k = tid; k < NPTS * 3; k += 256) P[k] = pb[k];
    __syncthreads();

    const int i = p0 + tid;
    const float xi = P[i * 3 + 0], yi = P[i * 3 + 1], zi = P[i * 3 + 2];

    float bd[NN_SIZE];
    int   bi[NN_SIZE];
    #pragma unroll
    for (int k = 0; k < NN_SIZE; ++k) { bd[k] = INFINITY; bi[k] = 0; }

    for (int j = 0; j < NPTS; ++j) {
        float dx = P[j * 3 + 0] - xi;
        float dy = P[j * 3 + 1] - yi;
        float dz = P[j * 3 + 2] - zi;
        float d2 = dx * dx + dy * dy + dz * dz;
        if (d2 < bd[NN_SIZE - 1]) {
            int k = NN_SIZE - 1;
            while (k > 0 && d2 < bd[k - 1]) {
                bd[k] = bd[k - 1]; bi[k] = bi[k - 1]; --k;
            }
            bd[k] = d2; bi[k] = j;
        }
    }

    float acc = 0.0f;
    #pragma unroll
    for (int k = 1; k < NN_SIZE; ++k) {   // drop self (slot 0)
        int j = bi[k];
        float dx = P[j * 3 + 0] - xi;
        float dy = P[j * 3 + 1] - yi;
        float dz = P[j * 3 + 2] - zi;
        float g2 = dx * dx + dy * dy + dz * dz;
        g2 = g2 < EPS_C ? EPS_C : g2;
        float dist = sqrtf(g2);
        acc += (RADIUS_C - dist) * expf(-g2 / (H_C * H_C));
    }

    red[tid] = acc;
    __syncthreads();
    for (int s = 128; s > 0; s >>= 1) {
        if (tid < s) red[tid] += red[tid + s];
        __syncthreads();
    }
    if (tid == 0) ws[WS_REP + blockIdx.x] = red[0];
}

// ---------------------------------------------------------------------------
// Kernel 4: finalize both scalars (fully overwrites d_out -> replay-safe)
// ---------------------------------------------------------------------------
__global__ void finalize_kernel(const float* __restrict__ ws,
                                const float* __restrict__ radius,
                                float* __restrict__ out) {
    float emd = 0.0f;
    for (int b = 0; b < NB; ++b) emd += ws[WS_DIST2 + b] / radius[b];
    emd = (emd / (float)NB) * 250.0f;

    float rep = 0.0f;
    for (int k = 0; k < 16; ++k) rep += ws[WS_REP + k];
    rep /= (float)(NB * NPTS * (NN_SIZE - 1));

    out[0] = emd;
    out[1] = rep;
}

extern "C" void kernel_launch(void* const* d_in, const int* in_sizes, int n_in,
                              void* d_out, int out_size, void* d_ws, size_t ws_size,
                              hipStream_t stream) {
    const float* pred   = (const float*)d_in[0];   // (4,1024,3)
    const float* gt     = (const float*)d_in[1];   // (4,1024,3)
    const float* radius = (const float*)d_in[2];   // (4,1)
    float* ws  = (float*)d_ws;
    float* out = (float*)d_out;

    // 1) pairwise d2 via WMMA: 4 batches * 64*64 tiles = 16384 waves / 8 per block
    d2_wmma_kernel<<<2048, 256, 0, stream>>>(pred, gt, ws);
    // 2) auction matching + EMD partial, one workgroup per batch
    auction_kernel<<<NB, NPTS, 0, stream>>>(pred, gt, ws);
    // 3) KNN repulsion partials
    rep_kernel<<<16, 256, 0, stream>>>(pred, ws);
    // 4) finalize
    finalize_kernel<<<1, 1, 0, stream>>>(ws, radius, out);
}